// MemoryOnlyTitan_53455162966285
// MI455X (gfx1250) — compile-verified
//
#include <hip/hip_runtime.h>

// ---------------------------------------------------------------------------
// MemoryOnlyTitan, reduced form (state0 == 0 makes both attentions uniform):
//   out[b,t,:] = LN( ((sum_p pm + sum_s x[b,s,:]) / T) @ Wv + bv ) * gamma
//                + beta, then @ Wout + bout,  broadcast over all t.
// Dominant cost: 67MB read of x + 64MB broadcast store => ~5.6us @ 23.3TB/s.
// The two 16x512 @ 512x512 GEMMs use V_WMMA_F32_16X16X4_F32 (wave32).
// ---------------------------------------------------------------------------

typedef __attribute__((ext_vector_type(2))) float v2f;
typedef __attribute__((ext_vector_type(8))) float v8f;

#define B_ 4
#define S_ 8192
#define D_ 512
#define P_ 32
#define T_ (P_ + S_)                 // 8224
#define NSEG 64
#define ROWS_PER_SEG (S_ / NSEG)     // 128

// workspace layout (float offsets)
#define WS_PARTIAL 0                          // [B][NSEG][D]   = 131072
#define WS_CMEAN   (B_ * NSEG * D_)           // [16][D] padded =   8192
#define WS_MEANV   (WS_CMEAN + 16 * D_)       // [4][D]         =   2048
#define WS_NORM    (WS_MEANV + B_ * D_)       // [16][D] padded =   8192
#define WS_OUTROW  (WS_NORM + 16 * D_)        // [4][D]         =   2048
// total ~606 KB of ws

// ---- Phase A: per-segment column sums of x ---------------------------------
__global__ void colsum_partial(const float* __restrict__ x, float* __restrict__ ws) {
    const int seg = blockIdx.x;          // 0..NSEG-1
    const int b   = blockIdx.y;          // 0..B-1
    const int tid = threadIdx.x;         // 256 threads, each owns 2 columns
    const float2* px = reinterpret_cast<const float2*>(
        x + (size_t)b * S_ * D_ + (size_t)seg * ROWS_PER_SEG * D_);
    float2 acc; acc.x = 0.0f; acc.y = 0.0f;
#pragma unroll 4
    for (int r = 0; r < ROWS_PER_SEG; ++r) {
        float2 v = px[r * (D_ / 2) + tid];
        acc.x += v.x; acc.y += v.y;
    }
    float2* out = reinterpret_cast<float2*>(
        ws + WS_PARTIAL + (size_t)(b * NSEG + seg) * D_);
    out[tid] = acc;
}

// ---- Phase B: fold partials + persistent memory -> cmean (16xD, padded) ----
__global__ void finalize_cmean(const float* __restrict__ pm, float* __restrict__ ws) {
    const int b   = blockIdx.x;          // 0..15 (rows 4..15 zero-padded)
    const int tid = threadIdx.x;         // 256 threads
    float2* cmean = reinterpret_cast<float2*>(ws + WS_CMEAN + (size_t)b * D_);
    if (b >= B_) { float2 z; z.x = 0.0f; z.y = 0.0f; cmean[tid] = z; return; }
    float2 acc; acc.x = 0.0f; acc.y = 0.0f;
    for (int seg = 0; seg < NSEG; ++seg) {
        float2 v = reinterpret_cast<const float2*>(
            ws + WS_PARTIAL + (size_t)(b * NSEG + seg) * D_)[tid];
        acc.x += v.x; acc.y += v.y;
    }
    for (int p = 0; p < P_; ++p) {
        float2 v = reinterpret_cast<const float2*>(pm + (size_t)p * D_)[tid];
        acc.x += v.x; acc.y += v.y;
    }
    const float invT = 1.0f / (float)T_;
    acc.x *= invT; acc.y *= invT;
    cmean[tid] = acc;
}

// ---- Phase C1/C3: Y[0..3][n] = (A16 @ W)[0..3][n] + bias[n]  via WMMA ------
// One wave (32 lanes) per 16-column tile; 128 steps of 16x16x4 f32 WMMA.
// A-matrix 16x4 layout: lanes 0-15 -> M=lane, K=0..1; lanes 16-31 -> K=2..3.
// B-matrix 4x16 mirrored: lanes 0-15 -> N=lane, K=0..1; lanes 16-31 -> K=2..3.
// C/D 16x16: VGPR i, lanes 0-15 -> M=i, N=lane; lanes 16-31 -> M=i+8.
__global__ void wmma_matvec16(const float* __restrict__ A16,   // [16][D]
                              const float* __restrict__ W,     // [D][D] row-major
                              const float* __restrict__ bias,  // [D]
                              float* __restrict__ Y) {         // [4][D]
    const int lane  = threadIdx.x;        // 32 threads = 1 wave
    const int nbase = blockIdx.x * 16;    // 32 blocks cover D=512 columns
    const int mn    = lane & 15;
    const int koff  = (lane >> 4) << 1;   // 0 or 2
    v8f acc = {0.f, 0.f, 0.f, 0.f, 0.f, 0.f, 0.f, 0.f};
    for (int k0 = 0; k0 < D_; k0 += 4) {
        v2f a, bf;
        a.x  = A16[(size_t)mn * D_ + k0 + koff];
        a.y  = A16[(size_t)mn * D_ + k0 + koff + 1];
        bf.x = W[(size_t)(k0 + koff)     * D_ + nbase + mn];
        bf.y = W[(size_t)(k0 + koff + 1) * D_ + nbase + mn];
        acc = __builtin_amdgcn_wmma_f32_16x16x4_f32(
            /*neg_a=*/false, a, /*neg_b=*/false, bf,
            /*c_mod=*/(short)0, acc, /*reuse_a=*/false, /*reuse_b=*/false);
    }
    if (lane < 16) {                       // rows 0..7 live in lanes 0-15
        const float bvv = bias[nbase + mn];
#pragma unroll
        for (int i = 0; i < 4; ++i)        // only batch rows 0..3 are real
            Y[(size_t)i * D_ + nbase + mn] = acc[i] + bvv;
    }
}

// ---- Phase C2: LayerNorm rows 0..3, zero-pad rows 4..15 --------------------
__global__ void layernorm_rows(const float* __restrict__ meanv,  // [4][D]
                               const float* __restrict__ gamma,
                               const float* __restrict__ beta,
                               float* __restrict__ norm16) {     // [16][D]
    const int b   = blockIdx.x;           // 0..15
    const int tid = threadIdx.x;          // 256 threads, 2 elems each
    float2* out = reinterpret_cast<float2*>(norm16 + (size_t)b * D_);
    if (b >= B_) { float2 z; z.x = 0.0f; z.y = 0.0f; out[tid] = z; return; }
    __shared__ float s1[256], s2[256];
    float2 v = reinterpret_cast<const float2*>(meanv + (size_t)b * D_)[tid];
    s1[tid] = v.x + v.y;
    s2[tid] = v.x * v.x + v.y * v.y;
    __syncthreads();
    for (int off = 128; off > 0; off >>= 1) {
        if (tid < off) { s1[tid] += s1[tid + off]; s2[tid] += s2[tid + off]; }
        __syncthreads();
    }
    const float mu  = s1[0] * (1.0f / D_);
    float var = s2[0] * (1.0f / D_) - mu * mu;
    var = fmaxf(var, 0.0f);
    const float rstd = __frsqrt_rn(var + 1e-5f);
    float2 g  = reinterpret_cast<const float2*>(gamma)[tid];
    float2 be = reinterpret_cast<const float2*>(beta)[tid];
    float2 o;
    o.x = (v.x - mu) * rstd * g.x + be.x;
    o.y = (v.y - mu) * rstd * g.y + be.y;
    out[tid] = o;
}

// ---- Phase D: broadcast the 4 result rows across all T tokens (64MB) -------
__global__ void broadcast_out(const float* __restrict__ outrow,  // [4][D]
                              float* __restrict__ out) {         // [B][T][D]
    const float4* row4 = reinterpret_cast<const float4*>(outrow); // [4][128]
    float4* o4 = reinterpret_cast<float4*>(out);
    const size_t total  = (size_t)B_ * T_ * (D_ / 4);
    const size_t stride = (size_t)gridDim.x * blockDim.x;
    for (size_t idx = (size_t)blockIdx.x * blockDim.x + threadIdx.x;
         idx < total; idx += stride) {
        const size_t row = idx >> 7;            // D/4 = 128 float4 per row
        const int d4 = (int)(idx & 127);
        const int b  = (int)(row / (size_t)T_); // const-div -> mul/shift
        o4[idx] = row4[b * 128 + d4];
    }
}

extern "C" void kernel_launch(void* const* d_in, const int* in_sizes, int n_in,
                              void* d_out, int out_size, void* d_ws, size_t ws_size,
                              hipStream_t stream) {
    const float* x     = (const float*)d_in[0];
    const float* pm    = (const float*)d_in[1];
    // d_in[2]=Wk, d_in[3]=bk, d_in[6]=Wq, d_in[7]=bq are mathematically dead.
    const float* Wv    = (const float*)d_in[4];
    const float* bv    = (const float*)d_in[5];
    const float* gamma = (const float*)d_in[8];
    const float* beta  = (const float*)d_in[9];
    const float* Wout  = (const float*)d_in[10];
    const float* bout  = (const float*)d_in[11];
    float* ws  = (float*)d_ws;
    float* out = (float*)d_out;

    colsum_partial<<<dim3(NSEG, B_), 256, 0, stream>>>(x, ws);
    finalize_cmean<<<16, 256, 0, stream>>>(pm, ws);
    wmma_matvec16<<<32, 32, 0, stream>>>(ws + WS_CMEAN, Wv, bv, ws + WS_MEANV);
    layernorm_rows<<<16, 256, 0, stream>>>(ws + WS_MEANV, gamma, beta, ws + WS_NORM);
    wmma_matvec16<<<32, 32, 0, stream>>>(ws + WS_NORM, Wout, bout, ws + WS_OUTROW);
    broadcast_out<<<4096, 256, 0, stream>>>(ws + WS_OUTROW, out);
}